// MyDense_68350109548836
// MI455X (gfx1250) — compile-verified
//
#include <hip/hip_runtime.h>
#include <cstdint>

// ---------------- problem constants (fixed by the reference) ----------------
#define N1D     571
#define PADW    55
#define GRID    2043                       // 681 * 3
#define GG      (2043u * 2043u)            // 4,173,849
#define NBATCH  8
#define TOTAL   (8u * 2043u * 2043u)       // 33,390,792 (divisible by 4)
#define FMAXC   2.0e-15f

// native clang vector (NOT HIP_vector_type) so nontemporal builtins accept it
typedef float vf4 __attribute__((ext_vector_type(4)));

__device__ __constant__ float kTwoPi = 6.283185307179586476925286766559f;
__device__ __constant__ float kClight = 299792458.0f;
__device__ __constant__ float kW0 = 1.8836515673088533e+15f; // 2*pi*c/1e-6

// ---------------------------------------------------------------------------
// Kernel A: build per-batch (cos, sin) LUT over the 571 radial bins.
// phase(b, i) = phi_s[i] + w[b] * gd_s[i]
//   phi_s = 0.5*(phi + reverse(phi))
//   gd_s  = 0.5*(sig(gd) + reverse(sig(gd))),  sig(x)=FMAX*sigmoid(x)-FMAX/2
//   w[b]  = 2*pi*(c / wvll[b]) - w0           (real part of reference's w)
// Only 8*571 = 4568 sincos evaluations total (vs 33.4M in the naive form).
// ---------------------------------------------------------------------------
__global__ void __launch_bounds__(128)
lut_kernel(const float* __restrict__ phi, const float* __restrict__ gd,
           const float* __restrict__ wvll, float2* __restrict__ lut)
{
    const int b = blockIdx.x;
    const float w = kTwoPi * (kClight / wvll[b]) - kW0;
    for (int i = threadIdx.x; i < N1D; i += blockDim.x) {
        const float p  = 0.5f * (phi[i] + phi[N1D - 1 - i]);
        const float g0 = FMAXC / (1.0f + expf(-gd[i]))           - 0.5f * FMAXC;
        const float g1 = FMAXC / (1.0f + expf(-gd[N1D - 1 - i])) - 0.5f * FMAXC;
        const float gs = 0.5f * (g0 + g1);
        const float ph = p + w * gs;
        float s, c;
        sincosf(ph, &s, &c);
        lut[b * N1D + i] = make_float2(c, s);
    }
}

// ---------------------------------------------------------------------------
// Kernel B: streaming complex modulation, 4 complex elements / thread.
//   out[e] = (re[e] + i*im[e]) * (c + i*s)   with (c,s) from the LDS LUT,
//   or (1,0) in the zero-padded ring (phase == 0 there).
// LUT staged into LDS with gfx1250 async global->LDS copies (ASYNCcnt),
// drained with s_wait_asynccnt. Streams use non-temporal hints so the
// 534 MB pass does not thrash the 192 MB L2 (radial_idx stays resident).
// ---------------------------------------------------------------------------
__global__ void __launch_bounds__(256)
apply_kernel(const float* __restrict__ re, const float* __restrict__ im,
             const int* __restrict__ ridx, const float2* __restrict__ lut_g,
             float* __restrict__ out)
{
    __shared__ __align__(16) float2 slut[NBATCH * N1D];   // 36,544 bytes

    // ---- async-stage the LUT into LDS (16B per lane per issue) ----
    // LDS byte offset = low 32 bits of the flat shared-space pointer.
    const uint32_t lds_base = (uint32_t)(uintptr_t)(&slut[0]);
    const int nchunk = (NBATCH * N1D * (int)sizeof(float2)) / 16;  // 2284
    const uint64_t gbase = (uint64_t)(uintptr_t)lut_g;
    for (int k = threadIdx.x; k < nchunk; k += 256) {
        const uint32_t loff = lds_base + (uint32_t)k * 16u;
        const uint64_t ga   = gbase + (uint64_t)k * 16u;
        asm volatile("global_load_async_to_lds_b128 %0, %1, off"
                     :: "v"(loff), "v"(ga) : "memory");
    }
    asm volatile("s_wait_asynccnt 0" ::: "memory");
    __syncthreads();

    // ---- stream 4 complex elements ----
    const uint32_t e0 = (blockIdx.x * 256u + threadIdx.x) * 4u;
    if (e0 >= TOTAL) return;

    const vf4 r4 = __builtin_nontemporal_load((const vf4*)(re + e0));
    const vf4 i4 = __builtin_nontemporal_load((const vf4*)(im + e0));
    const float rr[4] = { r4.x, r4.y, r4.z, r4.w };
    const float ii[4] = { i4.x, i4.y, i4.z, i4.w };

    // decompose flat index -> (b, y, x), then increment with carries
    uint32_t b   = e0 / GG;
    uint32_t rem = e0 - b * GG;
    uint32_t y   = rem / GRID;
    uint32_t x   = rem - y * GRID;

    float oc[8];
#pragma unroll
    for (int j = 0; j < 4; ++j) {
        const uint32_t iy = (y / 3u) - PADW;   // unsigned wrap => big if in pad
        const uint32_t ix = (x / 3u) - PADW;
        float cs_c = 1.0f, cs_s = 0.0f;
        if (iy < (uint32_t)N1D && ix < (uint32_t)N1D) {
            const int id = ridx[iy * N1D + ix];
            const float2 cs = slut[b * N1D + id];   // ds_load_b64
            cs_c = cs.x; cs_s = cs.y;
        }
        oc[2 * j]     = rr[j] * cs_c - ii[j] * cs_s;
        oc[2 * j + 1] = rr[j] * cs_s + ii[j] * cs_c;
        if (++x == GRID) { x = 0u; if (++y == GRID) { y = 0u; ++b; } }
    }

    float* op = out + 2u * (size_t)e0;
    vf4 o0; o0.x = oc[0]; o0.y = oc[1]; o0.z = oc[2]; o0.w = oc[3];
    vf4 o1; o1.x = oc[4]; o1.y = oc[5]; o1.z = oc[6]; o1.w = oc[7];
    __builtin_nontemporal_store(o0, (vf4*)op);
    __builtin_nontemporal_store(o1, (vf4*)(op + 4));
}

// ---------------------------------------------------------------------------
extern "C" void kernel_launch(void* const* d_in, const int* in_sizes, int n_in,
                              void* d_out, int out_size, void* d_ws, size_t ws_size,
                              hipStream_t stream)
{
    (void)in_sizes; (void)n_in; (void)out_size; (void)ws_size;
    const float* phi  = (const float*)d_in[0];   // [571]
    const float* gd   = (const float*)d_in[1];   // [571]
    const float* re   = (const float*)d_in[2];   // [8,2043,2043]
    const float* im   = (const float*)d_in[3];   // [8,2043,2043]
    const float* wvll = (const float*)d_in[4];   // [8]
    const int*   ridx = (const int*)d_in[5];     // [571,571]
    float2* lut = (float2*)d_ws;                 // 8*571 float2 = 36,544 B
    float*  out = (float*)d_out;                 // complex64 interleaved

    lut_kernel<<<NBATCH, 128, 0, stream>>>(phi, gd, wvll, lut);

    const uint32_t nthreads = TOTAL / 4u;                 // 8,347,698
    const uint32_t nblocks  = (nthreads + 255u) / 256u;   // 32,609
    apply_kernel<<<nblocks, 256, 0, stream>>>(re, im, ridx, lut, out);
}